// ODERNN_46926812677057
// MI455X (gfx1250) — compile-verified
//
#include <hip/hip_runtime.h>
#include <math.h>

// ---------------- problem constants (match reference) ----------------
#define BB    4
#define VV    2048
#define TT    12
#define HH    64
#define EE    32768
#define VT    (BB * VV)        // 8192
#define TOTK  27
#define KROW  (TOTK * HH)      // 1728
#define LDSPITCH 68            // padded row stride (floats) for A tile in LDS

typedef float v2f __attribute__((ext_vector_type(2)));
typedef float v8f __attribute__((ext_vector_type(8)));

enum { ACT_NONE = 0, ACT_ELU = 1, ACT_TANH_ADD = 2, ACT_SIG_ADD = 3, ACT_GRU = 4 };

// ---------------- spline basis: degree-1 open B-spline, DIM=3, K=3 ----------------
__global__ __launch_bounds__(256) void basis_kernel(const float* __restrict__ attr,
                                                    float* __restrict__ basis,
                                                    int* __restrict__ wi) {
    int e = blockIdx.x * 256 + threadIdx.x;
    if (e >= EE) return;
    float fr[3]; int lo[3];
#pragma unroll
    for (int d = 0; d < 3; ++d) {
        float u = attr[e * 3 + d] * 2.0f;              // * (K-1)
        float fl = floorf(u);
        fl = fminf(fmaxf(fl, 0.0f), 1.0f);             // clip to [0, K-2]
        fr[d] = u - fl;
        lo[d] = (int)fl;
    }
#pragma unroll
    for (int s = 0; s < 8; ++s) {
        float b = 1.0f; int idx = 0; int p = 1;
#pragma unroll
        for (int d = 0; d < 3; ++d) {
            int bit = (s >> d) & 1;
            b *= bit ? fr[d] : (1.0f - fr[d]);
            idx += (lo[d] + bit) * p;
            p *= 3;
        }
        basis[e * 8 + s] = b;
        wi[e * 8 + s] = idx;
    }
}

// ---------------- CSR construction (per-graph, reused by all B copies) ----------------
__global__ __launch_bounds__(256) void deg_count_kernel(const int* __restrict__ ei,
                                                        int* __restrict__ degI) {
    int e = blockIdx.x * 256 + threadIdx.x;
    if (e < EE) atomicAdd(&degI[ei[EE + e]], 1);
}

__global__ void scan_kernel(const int* __restrict__ degI, int* __restrict__ rowp) {
    if (threadIdx.x == 0 && blockIdx.x == 0) {
        int s = 0;
        for (int v = 0; v < VV; ++v) { rowp[v] = s; s += degI[v]; }
        rowp[VV] = s;
    }
}

__global__ __launch_bounds__(256) void csr_fill_kernel(const int* __restrict__ ei,
                                                       const int* __restrict__ rowp,
                                                       int* __restrict__ cursor,
                                                       int* __restrict__ elist) {
    int e = blockIdx.x * 256 + threadIdx.x;
    if (e < EE) {
        int d = ei[EE + e];
        int pos = atomicAdd(&cursor[d], 1);
        elist[rowp[d] + pos] = e;
    }
}

__global__ __launch_bounds__(256) void dinv_kernel(const int* __restrict__ degI,
                                                   float* __restrict__ dinv) {
    int n = blockIdx.x * 256 + threadIdx.x;
    if (n < VT) dinv[n] = 1.0f / fmaxf((float)degI[n & (VV - 1)], 1.0f);
}

// ---------------- weight pair-packing: (even-K,odd-K) rows -> contiguous float2 ------
// Wre[ci][k2][n] = { Wk[ci][2*k2][n], Wk[ci][2*k2+1][n] }  -> single b64 load per WMMA
__global__ __launch_bounds__(256) void reorder_wk_kernel(const float* __restrict__ Wk,
                                                         v2f* __restrict__ Wre) {
    int i = blockIdx.x * 256 + threadIdx.x;       // over 7*(KROW/2)*HH
    int n = i & (HH - 1);
    int rest = i >> 6;
    int k2 = rest % (KROW / 2);
    int ci = rest / (KROW / 2);
    const float* src = Wk + ((size_t)ci * KROW + 2 * k2) * HH + n;
    v2f pr; pr.x = src[0]; pr.y = src[HH];
    Wre[i] = pr;
}

__global__ __launch_bounds__(256) void reorder_wr_kernel(const float* __restrict__ Wroot,
                                                         v2f* __restrict__ Wre) {
    int i = blockIdx.x * 256 + threadIdx.x;       // over 7*(HH/2)*HH
    int n = i & (HH - 1);
    int rest = i >> 6;
    int k2 = rest % (HH / 2);
    int ci = rest / (HH / 2);
    const float* src = Wroot + ((size_t)ci * HH + 2 * k2) * HH + n;
    v2f pr; pr.x = src[0]; pr.y = src[HH];
    Wre[i] = pr;
}

// ---------------- CSR accumulate: Fv[node] built in LDS, no atomics, no memset -------
// One 64-thread block per destination node; thread c exclusively owns channel column c
// of the 27x64 tile, so the LDS read-modify-writes need no synchronization at all.
// feat element (n,c) = feat[n*fstride + foff + c]  (lets us read x time-slices in place)
__global__ __launch_bounds__(64) void accum_kernel(const float* __restrict__ feat,
                                                   int fstride, int foff,
                                                   const int* __restrict__ ei,
                                                   const float* __restrict__ basis,
                                                   const int* __restrict__ wi,
                                                   const int* __restrict__ rowp,
                                                   const int* __restrict__ elist,
                                                   float* __restrict__ Fv) {
    __shared__ float acc[TOTK * HH];              // 6912 B
    const int c = threadIdx.x;                    // channel column 0..63
    const int node = blockIdx.x;                  // 0..VT-1
    const int v = node & (VV - 1);
    const int b = node >> 11;                     // VV = 2^11
    const int beg = rowp[v], end = rowp[v + 1];

#pragma unroll
    for (int k = 0; k < TOTK; ++k) acc[k * HH + c] = 0.f;

    for (int i = beg; i < end; ++i) {
        int e = elist[i];                         // uniform across block -> scalar loads
        int src = ei[e] + b * VV;
        float xv = feat[(size_t)src * fstride + foff + c];
#pragma unroll
        for (int s = 0; s < 8; ++s) {
            float w = basis[e * 8 + s];
            int k = wi[e * 8 + s];
            acc[k * HH + c] += w * xv;
        }
    }

    float* outp = Fv + (size_t)node * KROW + c;
#pragma unroll
    for (int k = 0; k < TOTK; ++k) outp[(size_t)k * HH] = acc[k * HH + c];
}

// ---------------- WMMA fp32 GEMM + fused epilogue ------------------------------------
// out_raw = (Fv . Wk) * deg_inv + feat . Wroot + bias ; then MODE-specific epilogue.
// 4 waves/block share an LDS-staged 16-row A tile; V_WMMA_F32_16X16X4_F32 K loop;
// pair-packed B gives one coalesced global_load_b64 per WMMA.
template <int MODE>
__global__ __launch_bounds__(128) void spline_gemm_kernel(
        const float* __restrict__ Fv,
        const float* __restrict__ feat, int fstride, int foff,
        const v2f* __restrict__ Wre,      // [KROW/2][HH] pair-packed conv weights
        const v2f* __restrict__ Wrr,      // [HH/2][HH] pair-packed root weights
        const float* __restrict__ bias,
        const float* __restrict__ dinv,
        float* __restrict__ out,
        const float* __restrict__ aux0, const float* __restrict__ aux1,
        const float* __restrict__ aux2, const float* __restrict__ aux3,
        float* __restrict__ gout, int t) {
    __shared__ float atile[16 * LDSPITCH];        // 4352 B
    const int tid = threadIdx.x;
    const int lane = tid & 31;
    const int wave = tid >> 5;                    // 0..3 -> N tile
    const int m0 = blockIdx.x * 16;
    const int n0 = wave * 16;
    const int mm = lane & 15;                     // A: M=lane%16 ; B/C/D: N=lane%16
    const int kk = (lane >> 4) * 2;               // A/B: VGPR pair holds K=kk,kk+1
    const int ncol = n0 + mm;

    // cooperative A-tile loader mapping: 128 threads x 8 floats = 16x64 chunk
    const int lrow = tid >> 3;
    const int lcol = (tid & 7) * 8;
    const float* arow_g = Fv + (size_t)(m0 + lrow) * KROW + lcol;
    float* srow = &atile[lrow * LDSPITCH + lcol];

    const float* afrag = &atile[mm * LDSPITCH + kk];
    const float* frow = feat + (size_t)(m0 + mm) * fstride + foff + kk;  // root A rows
    const v2f* wkb = Wre + (size_t)(kk >> 1) * HH + ncol;
    const v2f* wrb = Wrr + (size_t)(kk >> 1) * HH + ncol;

    v8f acc = {0.f, 0.f, 0.f, 0.f, 0.f, 0.f, 0.f, 0.f};
    for (int k0 = 0; k0 < KROW; k0 += 64) {
        const float4* g = (const float4*)(arow_g + k0);
        float4 a0 = g[0], a1 = g[1];
        __syncthreads();
        ((float4*)srow)[0] = a0;
        ((float4*)srow)[1] = a1;
        __syncthreads();
#pragma unroll 4
        for (int k = 0; k < 64; k += 4) {
            v2f a = *(const v2f*)(afrag + k);
            v2f b = wkb[(size_t)((k0 + k) >> 1) * HH];
            acc = __builtin_amdgcn_wmma_f32_16x16x4_f32(false, a, false, b,
                                                        (short)0, acc, false, false);
        }
    }

    v8f acc2 = {0.f, 0.f, 0.f, 0.f, 0.f, 0.f, 0.f, 0.f};
#pragma unroll
    for (int k = 0; k < HH; k += 4) {
        v2f a = *(const v2f*)(frow + k);
        v2f b = wrb[(size_t)(k >> 1) * HH];
        acc2 = __builtin_amdgcn_wmma_f32_16x16x4_f32(false, a, false, b,
                                                     (short)0, acc2, false, false);
    }

    // C/D layout: VGPR r -> M = r + (lane<16 ? 0 : 8), N = lane%16
    const float bval = bias[ncol];
    const int mbase = m0 + ((lane >> 4) << 3);
#pragma unroll
    for (int r = 0; r < 8; ++r) {
        const int mrow = mbase + r;
        const size_t idx = (size_t)mrow * HH + ncol;
        float val = acc[r] * dinv[mrow] + acc2[r] + bval;
        if (MODE == ACT_NONE) {
            out[idx] = val;
        } else if (MODE == ACT_ELU) {
            out[idx] = val > 0.f ? val : (expf(val) - 1.f);
        } else if (MODE == ACT_TANH_ADD) {            // out = aux0 + tanh(val)
            out[idx] = aux0[idx] + tanhf(val);
        } else if (MODE == ACT_SIG_ADD) {             // out = sigmoid(val + aux0)
            out[idx] = 1.0f / (1.0f + expf(-(val + aux0[idx])));
        } else {                                      // GRU: aux = r, hr, z, ho
            float nn = tanhf(val + aux0[idx] * aux1[idx]);
            float zz = aux2[idx];
            float hnew = (1.0f - zz) * nn + zz * aux3[idx];
            out[idx] = hnew;                          // carried state h
            gout[((size_t)mrow * TT + t) * HH + ncol] = hnew;   // [B,V,T,H] slice
        }
    }
}

// ---------------- host orchestration ----------------
extern "C" void kernel_launch(void* const* d_in, const int* in_sizes, int n_in,
                              void* d_out, int out_size, void* d_ws, size_t ws_size,
                              hipStream_t stream) {
    const float* x         = (const float*)d_in[0];   // [B,V,T,H]
    const float* edge_attr = (const float*)d_in[1];   // [E,3]
    const float* Wk        = (const float*)d_in[2];   // [7,27,H,H]
    const float* Wroot     = (const float*)d_in[3];   // [7,H,H]
    const float* bias      = (const float*)d_in[4];   // [7,H]
    const int*   ei        = (const int*)d_in[5];     // [2,E]
    float* out = (float*)d_out;

    char* p = (char*)d_ws;
    auto alloc = [&](size_t bytes) -> void* {
        void* r = (void*)p;
        p += (bytes + 255) & ~(size_t)255;
        return r;
    };
    float* basisB = (float*)alloc((size_t)EE * 8 * sizeof(float));
    int*   wiB    = (int*)  alloc((size_t)EE * 8 * sizeof(int));
    int*   degI   = (int*)  alloc((size_t)VV * sizeof(int));
    int*   rowp   = (int*)  alloc((size_t)(VV + 1) * sizeof(int));
    int*   cursor = (int*)  alloc((size_t)VV * sizeof(int));
    int*   elist  = (int*)  alloc((size_t)EE * sizeof(int));
    float* dinv   = (float*)alloc((size_t)VT * sizeof(float));
    v2f*   WreK   = (v2f*)  alloc((size_t)7 * (KROW / 2) * HH * sizeof(v2f));
    v2f*   WreR   = (v2f*)  alloc((size_t)7 * (HH / 2) * HH * sizeof(v2f));
    float* Fv     = (float*)alloc((size_t)VT * KROW * sizeof(float));
    const size_t NH = (size_t)VT * HH;
    float* h  = (float*)alloc(NH * sizeof(float));
    float* ho = (float*)alloc(NH * sizeof(float));
    float* hr = (float*)alloc(NH * sizeof(float));
    float* rb = (float*)alloc(NH * sizeof(float));
    float* zb = (float*)alloc(NH * sizeof(float));
    float* ta = (float*)alloc(NH * sizeof(float));
    float* t1 = (float*)alloc(NH * sizeof(float));

    const int gmBlocks = VT / 16;                     // gemm grid (512)

    // ---- static prep: basis + CSR + deg_inv + weight pair-packing ----
    basis_kernel<<<EE / 256, 256, 0, stream>>>(edge_attr, basisB, wiB);
    hipMemsetAsync(degI, 0, (size_t)VV * sizeof(int), stream);
    hipMemsetAsync(cursor, 0, (size_t)VV * sizeof(int), stream);
    deg_count_kernel<<<EE / 256, 256, 0, stream>>>(ei, degI);
    scan_kernel<<<1, 1, 0, stream>>>(degI, rowp);
    csr_fill_kernel<<<EE / 256, 256, 0, stream>>>(ei, rowp, cursor, elist);
    dinv_kernel<<<VT / 256, 256, 0, stream>>>(degI, dinv);
    reorder_wk_kernel<<<(7 * (KROW / 2) * HH) / 256, 256, 0, stream>>>(Wk, WreK);
    reorder_wr_kernel<<<(7 * (HH / 2) * HH) / 256, 256, 0, stream>>>(Wroot, WreR);
    hipMemsetAsync(h, 0, NH * sizeof(float), stream);

    auto accum = [&](const float* feat, int fstride, int foff) {
        accum_kernel<<<VT, 64, 0, stream>>>(feat, fstride, foff,
                                            ei, basisB, wiB, rowp, elist, Fv);
    };
    const v2f* WKc[7]; const v2f* WRc[7]; const float* Bc[7];
    for (int ci = 0; ci < 7; ++ci) {
        WKc[ci] = WreK + (size_t)ci * (KROW / 2) * HH;
        WRc[ci] = WreR + (size_t)ci * (HH / 2) * HH;
        Bc[ci]  = bias + (size_t)ci * HH;
    }

    for (int t = 0; t < TT; ++t) {
        const int xoff = t * HH;                  // x time-slice: stride TT*HH
        // ---- ODE Euler: ho = h + tanh(conv6(elu(conv5(h)))) ----
        accum(h, HH, 0);                          // Fv(h)
        spline_gemm_kernel<ACT_ELU><<<gmBlocks, 128, 0, stream>>>(
            Fv, h, HH, 0, WKc[5], WRc[5], Bc[5], dinv, ta,
            nullptr, nullptr, nullptr, nullptr, nullptr, 0);
        accum(ta, HH, 0);                         // Fv(ta)
        spline_gemm_kernel<ACT_TANH_ADD><<<gmBlocks, 128, 0, stream>>>(
            Fv, ta, HH, 0, WKc[6], WRc[6], Bc[6], dinv, ho,
            h, nullptr, nullptr, nullptr, nullptr, 0);
        // ---- GRU: one accum feeds conv1+conv3 (ho); one feeds conv0+conv2+conv4 (xt)
        accum(ho, HH, 0);                         // Fv(ho)
        spline_gemm_kernel<ACT_NONE><<<gmBlocks, 128, 0, stream>>>(
            Fv, ho, HH, 0, WKc[1], WRc[1], Bc[1], dinv, hr,
            nullptr, nullptr, nullptr, nullptr, nullptr, 0);
        spline_gemm_kernel<ACT_NONE><<<gmBlocks, 128, 0, stream>>>(
            Fv, ho, HH, 0, WKc[3], WRc[3], Bc[3], dinv, t1,
            nullptr, nullptr, nullptr, nullptr, nullptr, 0);
        accum(x, TT * HH, xoff);                  // Fv(xt)
        spline_gemm_kernel<ACT_SIG_ADD><<<gmBlocks, 128, 0, stream>>>(
            Fv, x, TT * HH, xoff, WKc[0], WRc[0], Bc[0], dinv, rb,
            hr, nullptr, nullptr, nullptr, nullptr, 0);      // r = sig(c0 + hr)
        spline_gemm_kernel<ACT_SIG_ADD><<<gmBlocks, 128, 0, stream>>>(
            Fv, x, TT * HH, xoff, WKc[2], WRc[2], Bc[2], dinv, zb,
            t1, nullptr, nullptr, nullptr, nullptr, 0);      // z = sig(c2 + c3)
        spline_gemm_kernel<ACT_GRU><<<gmBlocks, 128, 0, stream>>>(
            Fv, x, TT * HH, xoff, WKc[4], WRc[4], Bc[4], dinv, h,
            rb, hr, zb, ho, out, t);              // h' = (1-z)*tanh(c4+r*hr)+z*ho
    }
}